// GPPrediction_13855564497138
// MI455X (gfx1250) — compile-verified
//
#include <hip/hip_runtime.h>

typedef float v2f __attribute__((ext_vector_type(2)));
typedef float v8f __attribute__((ext_vector_type(8)));

#define CTX   448      // context length (T - N_PREDICTORS)
#define NPRED 64       // predictors (M of the GEMM)
#define TLEN  512
#define DDIM  1024
#define NB    64       // batch
#define AUGC  512      // augmented columns: 448 (A) + 64 (k_pred RHS)
#define KSTR  450      // padded LDS row stride for k_mult (448 % 64 == 0 -> pad)

// ---------------------------------------------------------------------------
// Kernel 1: build augmented [A | k_pred], row-major 448 x 512 in workspace.
// t[i] = i/511 * 0.3 ;  K[i,j] = exp(-0.5*(t_i - t_j)^2 / 0.01) ; +sigma on diag.
// Note t_pred index 448+(j-448) == j, so t_j = j*scale uniformly.
// ---------------------------------------------------------------------------
__global__ void gp_build(float* __restrict__ aug) {
  int idx = blockIdx.x * blockDim.x + threadIdx.x;
  if (idx >= CTX * AUGC) return;
  int i = idx / AUGC;
  int j = idx % AUGC;
  const float sc = 0.3f / 511.0f;
  float d = (i - j) * sc;
  float v = __expf(-50.0f * d * d);      // -0.5/L = -50
  if (j == i) v += 0.01f;                // + SIGMA (only hits the cov block)
  aug[idx] = v;
}

// ---------------------------------------------------------------------------
// Kernel 2: single-workgroup Gaussian elimination (A is SPD -> no pivoting)
// on [A | B] followed by upward Jordan sweep on the 64 RHS columns.
// Leaves X = inv(A) @ k_pred in columns 448..511.
// volatile -> coherent-scope global ops between barriers.
// ---------------------------------------------------------------------------
__global__ void __launch_bounds__(512) gp_solve(float* aug_) {
  volatile float* aug = aug_;
  __shared__ float prow[AUGC];
  __shared__ float pscale;
  const int tid = threadIdx.x;

  // forward elimination
  for (int k = 0; k < CTX; ++k) {
    if (tid == 0) pscale = 1.0f / aug[k * AUGC + k];
    prow[tid] = aug[k * AUGC + tid];           // stage pivot row in LDS
    __syncthreads();
    const float ip = pscale;
    for (int i = k + 1 + tid; i < CTX; i += 512) {   // one row per thread
      float f = aug[i * AUGC + k] * ip;
      for (int j = k + 1; j < AUGC; ++j)
        aug[i * AUGC + j] -= f * prow[j];
      aug[i * AUGC + k] = 0.0f;
    }
    __syncthreads();
  }

  // upward (Jordan) elimination restricted to the RHS block
  for (int k = CTX - 1; k >= 0; --k) {
    if (tid < NPRED) {
      float x = aug[k * AUGC + CTX + tid] / aug[k * AUGC + k];
      aug[k * AUGC + CTX + tid] = x;
      prow[tid] = x;
    }
    __syncthreads();
    for (int i = tid; i < k; i += 512) {
      float f = aug[i * AUGC + k];
      #pragma unroll 8
      for (int c = 0; c < NPRED; ++c)
        aug[i * AUGC + CTX + c] -= f * prow[c];
    }
    __syncthreads();
  }
}

// ---------------------------------------------------------------------------
// Kernel 3: batched GEMM  out[b,p,d] = sum_c kmult[p,c] * z[b,c,d]
// via V_WMMA_F32_16X16X4_F32.  Block = 256 thr (8 waves) computes a full
// 64(M) x 64(N) tile for one batch; wave = 16x32 (two 16x16 f32 accums).
// k_mult staged in LDS (read transposed out of the solver's RHS columns).
// ---------------------------------------------------------------------------
__global__ void __launch_bounds__(256) gp_gemm(const float* __restrict__ z,
                                               const float* __restrict__ aug,
                                               float* __restrict__ out) {
  extern __shared__ float km[];            // [NPRED][KSTR]
  const int b    = blockIdx.y;
  const int d0   = blockIdx.x * 64;
  const int tid  = threadIdx.x;
  const int lane = tid & 31;
  const int wave = tid >> 5;
  const int m0   = (wave >> 1) * 16;       // 4 M-waves
  const int n0   = d0 + (wave & 1) * 32;   // 2 N-waves

  // stage k_mult[p][c] = X[c][p] : coalesced global reads, padded LDS rows
  for (int idx = tid; idx < NPRED * CTX; idx += 256) {
    int c = idx >> 6;                      // 0..447
    int p = idx & 63;                      // 0..63
    km[p * KSTR + c] = aug[c * AUGC + CTX + p];
  }
  __syncthreads();

  // WMMA f32 16x16x4 fragment coordinates (wave32):
  //  A: lanes 0-15 rows M, VGPR{0,1}=K{0,1}; lanes 16-31 same rows, K{2,3}
  //  B: lanes 0-15 cols N, VGPR{0,1}=K{0,1}; lanes 16-31 cols, K{2,3}
  const int mrow  = m0 + (lane & 15);
  const int khalf = (lane >> 4) << 1;      // 0 or 2
  const int ncol  = n0 + (lane & 15);
  const float* zb = z + (size_t)b * (TLEN * DDIM);

  v8f acc0 = {0.f,0.f,0.f,0.f,0.f,0.f,0.f,0.f};
  v8f acc1 = acc0;

  #pragma unroll 4
  for (int k0 = 0; k0 < CTX; k0 += 4) {
    const int kk = k0 + khalf;
    v2f a = *(const v2f*)&km[mrow * KSTR + kk];          // ds_load_b64
    const float* zp = zb + (size_t)kk * DDIM + ncol;
    v2f b0, b1;
    b0.x = zp[0];        b0.y = zp[DDIM];
    b1.x = zp[16];       b1.y = zp[DDIM + 16];
    if (k0 + 34 < CTX) __builtin_prefetch(zp + 32 * DDIM, 0, 1);  // global_prefetch_b8
    acc0 = __builtin_amdgcn_wmma_f32_16x16x4_f32(false, a, false, b0,
                                                 (short)0, acc0, false, false);
    acc1 = __builtin_amdgcn_wmma_f32_16x16x4_f32(false, a, false, b1,
                                                 (short)0, acc1, false, false);
  }

  // C/D layout: VGPR v -> M = v (lanes 0-15) or v+8 (lanes 16-31)
  float* op = out + (size_t)b * NPRED * DDIM;
  const int mbase = m0 + ((lane >> 4) << 3);
  #pragma unroll
  for (int v = 0; v < 8; ++v) {
    const int m = mbase + v;
    op[(size_t)m * DDIM + ncol]      = acc0[v];
    op[(size_t)m * DDIM + ncol + 16] = acc1[v];
  }
}

// ---------------------------------------------------------------------------
extern "C" void kernel_launch(void* const* d_in, const int* in_sizes, int n_in,
                              void* d_out, int out_size, void* d_ws, size_t ws_size,
                              hipStream_t stream) {
  (void)in_sizes; (void)n_in; (void)out_size; (void)ws_size;
  const float* z   = (const float*)d_in[0];
  float*       out = (float*)d_out;
  float*       aug = (float*)d_ws;        // needs 448*512*4 = 917504 bytes

  gp_build<<<(CTX * AUGC + 255) / 256, 256, 0, stream>>>(aug);
  gp_solve<<<1, 512, 0, stream>>>(aug);
  gp_gemm<<<dim3(DDIM / 64, NB), 256, NPRED * KSTR * sizeof(float), stream>>>(z, aug, out);
}